// Gate_515396075782
// MI455X (gfx1250) — compile-verified
//
#include <hip/hip_runtime.h>
#include <hip/hip_bf16.h>
#include <math.h>

typedef float v2f __attribute__((ext_vector_type(2)));
typedef float v4f __attribute__((ext_vector_type(4)));
typedef float v8f __attribute__((ext_vector_type(8)));

#define B_SZ   64
#define C_IN   512
#define HW     3136          // 56*56
#define HC     16
#define OC     512
#define BN_EPS 1e-5f

// ---------------------------------------------------------------------------
// Kernel 1: global average pool  [64,512,56,56] -> [64,512]
// One 256-thread block per (b,c) channel; 3136 floats = 784 float4.
// Non-temporal b128 loads (411 MB read-once stream; don't thrash L2).
// ---------------------------------------------------------------------------
__global__ __launch_bounds__(256) void Gate_pool_kernel(
    const float* __restrict__ x, float* __restrict__ pooled) {
  const int bc = blockIdx.x;                       // 0 .. 64*512-1
  const v4f* xv = (const v4f*)(x + (size_t)bc * HW);
  const int t = threadIdx.x;

  float s = 0.0f;
#pragma unroll
  for (int i = 0; i < 3; ++i) {                    // 3*256 = 768 float4
    v4f v = __builtin_nontemporal_load(&xv[t + i * 256]);
    s += v.x + v.y + v.z + v.w;
  }
  if (t < 16) {                                    // remaining 16 float4
    v4f v = __builtin_nontemporal_load(&xv[768 + t]);
    s += v.x + v.y + v.z + v.w;
  }

  // wave32 reduction
#pragma unroll
  for (int off = 16; off > 0; off >>= 1)
    s += __shfl_xor(s, off, 32);

  __shared__ float part[8];
  const int wave = t >> 5, lane = t & 31;
  if (lane == 0) part[wave] = s;
  __syncthreads();
  if (t == 0) {
    float tot = 0.0f;
#pragma unroll
    for (int w = 0; w < 8; ++w) tot += part[w];
    pooled[bc] = tot * (1.0f / (float)HW);
  }
}

// ---------------------------------------------------------------------------
// Kernel 2: fused fc1 -> batchnorm(train) -> relu -> fc2(+bias) -> gumbel gate
// Single block, 4 wave32's. Each wave owns a 16-row batch tile.
// f32 WMMA 16x16x4 keeps the threshold decision bit-accurate in f32.
// ---------------------------------------------------------------------------
__global__ __launch_bounds__(128) void Gate_fused_kernel(
    const float* __restrict__ pooled,   // [64,512]
    const float* __restrict__ u,        // [64,512]
    const float* __restrict__ w1,       // [16,512]
    const float* __restrict__ gamma_,   // [16]
    const float* __restrict__ beta_,    // [16]
    const float* __restrict__ w2,       // [512,16]
    const float* __restrict__ b2,       // [512]
    float* __restrict__ out)            // [64,512]
{
  __shared__ float hbuf[B_SZ][HC];      // hidden activations
  __shared__ float scale_s[HC], shift_s[HC];

  const int t     = threadIdx.x;
  const int wave  = t >> 5;
  const int lane  = t & 31;
  const int mBase = wave * 16;          // batch-row tile per wave
  const int half  = lane >> 4;          // 0: lanes 0-15, 1: lanes 16-31
  const int mn    = lane & 15;          // M (for A) / N (for B,C,D)
  const int koff  = half * 2;           // K offset within 4-wide step

  // ---- Stage A: h = pooled @ w1^T  (M=16 per wave, N=16, K=512) ----
  v8f c = {};
  for (int k0 = 0; k0 < C_IN; k0 += 4) {
    const int k = k0 + koff;
    v2f a, b;
    a.x = pooled[(mBase + mn) * C_IN + k];
    a.y = pooled[(mBase + mn) * C_IN + k + 1];
    b.x = w1[mn * C_IN + k];            // B[k][n] = w1[n][k]
    b.y = w1[mn * C_IN + k + 1];
    c = __builtin_amdgcn_wmma_f32_16x16x4_f32(
        false, a, false, b, (short)0, c, false, false);
  }
#pragma unroll
  for (int r = 0; r < 8; ++r)           // D layout: (M = r + 8*half, N = mn)
    hbuf[mBase + r + half * 8][mn] = c[r];
  __syncthreads();

  // ---- Stage B: batchnorm over batch dim (training stats, ddof=0) ----
  if (t < HC) {
    float sum = 0.0f, sq = 0.0f;
    for (int bb = 0; bb < B_SZ; ++bb) {
      float v = hbuf[bb][t];
      sum += v; sq += v * v;
    }
    float mu  = sum * (1.0f / B_SZ);
    float var = sq * (1.0f / B_SZ) - mu * mu;
    float rs  = rsqrtf(var + BN_EPS) * gamma_[t];
    scale_s[t] = rs;
    shift_s[t] = beta_[t] - mu * rs;
  }
  __syncthreads();

  // normalize + relu in place
  for (int i = t; i < B_SZ * HC; i += 128) {
    const int bb = i >> 4, hh = i & 15;
    float v = hbuf[bb][hh] * scale_s[hh] + shift_s[hh];
    hbuf[bb][hh] = v > 0.0f ? v : 0.0f;
  }
  __syncthreads();

  // ---- Stage C: log_alpha = h @ w2^T + b2 ; gate = (log_alpha + l) > 0 ----
  for (int nt = 0; nt < OC / 16; ++nt) {
    const int nBase = nt * 16;
    v8f acc = {};
#pragma unroll
    for (int k0 = 0; k0 < HC; k0 += 4) {
      const int k = k0 + koff;
      v2f a, b;
      a.x = hbuf[mBase + mn][k];
      a.y = hbuf[mBase + mn][k + 1];
      b.x = w2[(nBase + mn) * HC + k];  // B[k][n] = w2[n][k]
      b.y = w2[(nBase + mn) * HC + k + 1];
      acc = __builtin_amdgcn_wmma_f32_16x16x4_f32(
          false, a, false, b, (short)0, acc, false, false);
    }
    const float bias = b2[nBase + mn];
#pragma unroll
    for (int r = 0; r < 8; ++r) {
      const int row = mBase + r + half * 8;
      const int col = nBase + mn;
      const float la = acc[r] + bias;
      const float uu = u[row * OC + col];
      const float l  = logf(uu) - log1pf(-uu);
      // sign((la + l)/T) == sign(la + l) since T > 0
      out[row * OC + col] = (la + l > 0.0f) ? 1.0f : 0.0f;
    }
  }
}

// ---------------------------------------------------------------------------
extern "C" void kernel_launch(void* const* d_in, const int* in_sizes, int n_in,
                              void* d_out, int out_size, void* d_ws, size_t ws_size,
                              hipStream_t stream) {
  const float* x      = (const float*)d_in[0];  // [64,512,56,56]
  const float* u      = (const float*)d_in[1];  // [64,512]
  const float* w1     = (const float*)d_in[2];  // [16,512]
  const float* gamma_ = (const float*)d_in[3];  // [16]
  const float* beta_  = (const float*)d_in[4];  // [16]
  const float* w2     = (const float*)d_in[5];  // [512,16]
  const float* b2     = (const float*)d_in[6];  // [512]
  float* out    = (float*)d_out;                // [64,512,1,1]
  float* pooled = (float*)d_ws;                 // [64,512] scratch (128 KB)

  Gate_pool_kernel<<<B_SZ * C_IN, 256, 0, stream>>>(x, pooled);
  Gate_fused_kernel<<<1, 128, 0, stream>>>(pooled, u, w1, gamma_, beta_,
                                           w2, b2, out);
}